// GNNModel_48842368090682
// MI455X (gfx1250) — compile-verified
//
#include <hip/hip_runtime.h>

typedef float v2f __attribute__((ext_vector_type(2)));
typedef float v8f __attribute__((ext_vector_type(8)));

#if defined(__gfx1250__) && __has_builtin(__builtin_amdgcn_wmma_f32_16x16x4_f32)
#define HAVE_WMMA_F32 1
#else
#define HAVE_WMMA_F32 0
#endif

// ---------------------------------------------------------------------------
// Utility: grid-stride zero fill
// ---------------------------------------------------------------------------
__global__ void gcn_zero(float* __restrict__ p, long n) {
    long i = (long)blockIdx.x * blockDim.x + threadIdx.x;
    const long stride = (long)gridDim.x * blockDim.x;
    for (; i < n; i += stride) p[i] = 0.0f;
}

// ---------------------------------------------------------------------------
// Degree counting: deg_out[src[e]] += 1, deg_in[dst[e]] += 1
// ---------------------------------------------------------------------------
__global__ void gcn_degrees(const int* __restrict__ src, const int* __restrict__ dst,
                            float* __restrict__ degO, float* __restrict__ degI, int E) {
    const long i = (long)blockIdx.x * blockDim.x + threadIdx.x;
    if (i < E) {
        atomicAdd(&degO[src[i]], 1.0f);
        atomicAdd(&degI[dst[i]], 1.0f);
    }
}

// deg -> clamp(deg,1)^-1/2, in place over both arrays (2N contiguous)
__global__ void gcn_invsqrt(float* __restrict__ p, long n) {
    const long i = (long)blockIdx.x * blockDim.x + threadIdx.x;
    if (i < n) {
        const float v = fmaxf(p[i], 1.0f);
        p[i] = 1.0f / sqrtf(v);
    }
}

// ---------------------------------------------------------------------------
// WMMA fp32 GEMM (K fixed at 64):  H[r, :] = rowscale[r] * (X[r, :] @ W)
//   X: [nrows, ldx] row-major
//   W: [64, wcols] row-major (columns >= wcols treated as 0 — padded in LDS)
//   H: [nrows, ldh] row-major, NCT*16 columns written
// One wave computes a 16-row strip of NCT 16x16 tiles, K stepped by 4 with
// V_WMMA_F32_16X16X4_F32 (fully unrolled: 16*NCT WMMAs per wave).
// W is staged TRANSPOSED + zero-padded in LDS so the inner loop is pure
// global_load_b64 (A) + ds_load_b64 (B) + wmma with uniform control flow.
// ---------------------------------------------------------------------------
template <int NCT>
__global__ __launch_bounds__(128)
void gcn_gemm_wmma64(const float* __restrict__ X, int ldx,
                     const float* __restrict__ W, int wcols,
                     const float* __restrict__ rowscale,
                     float* __restrict__ H, int ldh, int nrows) {
    constexpr int KD  = 64;
    constexpr int LDW = KD + 2;                 // 66-float pitch: staggers LDS banks
    __shared__ float sWt[NCT * 16 * LDW];       // W^T: [n][k], zero-padded cols

    // Cooperative fill: coalesced global reads, transposed LDS writes.
    for (int idx = threadIdx.x; idx < KD * NCT * 16; idx += (int)blockDim.x) {
        const int k = idx / (NCT * 16);
        const int n = idx - k * (NCT * 16);
        sWt[n * LDW + k] = (n < wcols) ? W[k * wcols + n] : 0.0f;
    }
    __syncthreads();

    const int lane = threadIdx.x & 31;
    const int wave = threadIdx.x >> 5;
    const int nrowtiles = (nrows + 15) >> 4;
    const int rowtile = blockIdx.x * (blockDim.x >> 5) + wave;
    if (rowtile >= nrowtiles) return;

    const int row0 = rowtile << 4;
    const int m  = lane & 15;   // row (A) / col (B,D) within tile
    const int kh = lane >> 4;   // K-half selector

    int rA = row0 + m;
    if (rA >= nrows) rA = nrows - 1;            // clamp keeps EXEC all-ones
    const float* aRow = X + (long)rA * ldx;

#if HAVE_WMMA_F32
    v8f acc[NCT];
#pragma unroll
    for (int t = 0; t < NCT; ++t)
        acc[t] = (v8f){0.f, 0.f, 0.f, 0.f, 0.f, 0.f, 0.f, 0.f};

#pragma unroll
    for (int k0 = 0; k0 < KD; k0 += 4) {
        const int ka = k0 + 2 * kh;             // this lane's K pair
        const v2f a = *reinterpret_cast<const v2f*>(aRow + ka);      // 8B load
#pragma unroll
        for (int t = 0; t < NCT; ++t) {
            const int n = t * 16 + m;
            const v2f b = *reinterpret_cast<const v2f*>(&sWt[n * LDW + ka]); // ds_load_b64
            acc[t] = __builtin_amdgcn_wmma_f32_16x16x4_f32(
                false, a, false, b, (short)0, acc[t], false, false);
        }
    }

#pragma unroll
    for (int t = 0; t < NCT; ++t) {
        const int col = t * 16 + m;
#pragma unroll
        for (int g = 0; g < 8; ++g) {
            const int r = row0 + g + 8 * kh;    // D layout: row = g + 8*kh
            if (r < nrows) {
                const float s = rowscale ? rowscale[r] : 1.0f;
                H[(long)r * ldh + col] = acc[t][g] * s;
            }
        }
    }
#else
    // Scalar fallback (also parsed by the host pass)
    for (int t = 0; t < NCT; ++t) {
        const int col = t * 16 + m;
        for (int g = 0; g < 8; ++g) {
            const int r = row0 + g + 8 * kh;
            if (r >= nrows) continue;
            float acc = 0.0f;
            for (int k = 0; k < KD; ++k)
                acc += X[(long)r * ldx + k] * sWt[col * LDW + k];
            const float s = rowscale ? rowscale[r] : 1.0f;
            H[(long)r * ldh + col] = acc * s;
        }
    }
#endif
}

// Generic scalar GEMM (only used if kdim != 64 — never in this workload)
__global__ void gcn_gemm_any(const float* __restrict__ X, int ldx, int kdim,
                             const float* __restrict__ W, int wcols,
                             const float* __restrict__ rowscale,
                             float* __restrict__ H, int ldh, int ncols, int nrows) {
    const long idx = (long)blockIdx.x * blockDim.x + threadIdx.x;
    const long total = (long)nrows * ncols;
    if (idx >= total) return;
    const int r = (int)(idx / ncols);
    const int col = (int)(idx - (long)r * ncols);
    float acc = 0.0f;
    for (int k = 0; k < kdim; ++k) {
        const float wv = (col < wcols) ? W[(long)k * wcols + col] : 0.0f;
        acc += X[(long)r * ldx + k] * wv;
    }
    H[(long)r * ldh + col] = acc * (rowscale ? rowscale[r] : 1.0f);
}

// ---------------------------------------------------------------------------
// Edge aggregation: M[dst[e], :] += H[src[e], :]
// Thread handles one (edge, 4-feature group).  16 (or 10) consecutive threads
// share an edge -> coalesced 256B row gathers out of L2, float atomics to L2.
// ---------------------------------------------------------------------------
template <int NG>
__global__ __launch_bounds__(256)
void gcn_aggregate(const float* __restrict__ H, int ldh,
                   const int* __restrict__ src, const int* __restrict__ dst,
                   float* __restrict__ M, int ldm, int nedges) {
    const long idx = (long)blockIdx.x * blockDim.x + threadIdx.x;
    const long total = (long)nedges * NG;
    if (idx >= total) return;
    const int e = (int)(idx / NG);
    const int g = (int)(idx - (long)e * NG);
    const int s = src[e];
    const int d = dst[e];
    const float4 v = *reinterpret_cast<const float4*>(H + (long)s * ldh + g * 4);
    float* out = M + (long)d * ldm + g * 4;
    atomicAdd(out + 0, v.x);
    atomicAdd(out + 1, v.y);
    atomicAdd(out + 2, v.z);
    atomicAdd(out + 3, v.w);
}

// Generic (non-multiple-of-4 feature count) fallback
__global__ void gcn_aggregate_any(const float* __restrict__ H, int ldh,
                                  const int* __restrict__ src, const int* __restrict__ dst,
                                  float* __restrict__ M, int ldm, int nfeat, int nedges) {
    const long idx = (long)blockIdx.x * blockDim.x + threadIdx.x;
    const long total = (long)nedges * nfeat;
    if (idx >= total) return;
    const int e = (int)(idx / nfeat);
    const int f = (int)(idx - (long)e * nfeat);
    atomicAdd(M + (long)dst[e] * ldm + f, H[(long)src[e] * ldh + f]);
}

// ---------------------------------------------------------------------------
// Epilogue: M[r,c] = M[r,c] * norm_in[r] + bias[c]  (optional ReLU), in place.
// M is dense [nrows, nfeat] (ld == nfeat).
// ---------------------------------------------------------------------------
__global__ void gcn_finalize(float* __restrict__ M, int nfeat,
                             const float* __restrict__ nrmI,
                             const float* __restrict__ bias,
                             int nrows, int doRelu) {
    const long idx = (long)blockIdx.x * blockDim.x + threadIdx.x;
    const long total = (long)nrows * nfeat;
    if (idx >= total) return;
    const int r = (int)(idx / nfeat);
    const int c = (int)(idx - (long)r * nfeat);
    float v = M[idx] * nrmI[r] + bias[c];
    if (doRelu) v = fmaxf(v, 0.0f);
    M[idx] = v;
}

// ---------------------------------------------------------------------------
// Host-side dispatch helpers
// ---------------------------------------------------------------------------
static void launch_gemm(int nct, int kdim, const float* X, int ldx,
                        const float* W, int wcols, const float* rowscale,
                        float* H, int ldh, int nrows, hipStream_t stream) {
    if (kdim == 64) {
        const int nrowtiles = (nrows + 15) / 16;
        const unsigned grid = (unsigned)((nrowtiles + 3) / 4);   // 4 waves / block
        switch (nct) {
            case 1: gcn_gemm_wmma64<1><<<grid, 128, 0, stream>>>(X, ldx, W, wcols, rowscale, H, ldh, nrows); break;
            case 2: gcn_gemm_wmma64<2><<<grid, 128, 0, stream>>>(X, ldx, W, wcols, rowscale, H, ldh, nrows); break;
            case 3: gcn_gemm_wmma64<3><<<grid, 128, 0, stream>>>(X, ldx, W, wcols, rowscale, H, ldh, nrows); break;
            default: gcn_gemm_wmma64<4><<<grid, 128, 0, stream>>>(X, ldx, W, wcols, rowscale, H, ldh, nrows); break;
        }
    } else {
        const int ncols = nct * 16;
        const long total = (long)nrows * ncols;
        gcn_gemm_any<<<(unsigned)((total + 255) / 256), 256, 0, stream>>>(
            X, ldx, kdim, W, wcols, rowscale, H, ldh, ncols, nrows);
    }
}

static void launch_aggregate(const float* H, int ldh, const int* src, const int* dst,
                             float* M, int ldm, int nfeat, int E, hipStream_t stream) {
    if (nfeat == 64) {
        const long total = (long)E * 16;
        gcn_aggregate<16><<<(unsigned)((total + 255) / 256), 256, 0, stream>>>(H, ldh, src, dst, M, ldm, E);
    } else if (nfeat == 40) {
        const long total = (long)E * 10;
        gcn_aggregate<10><<<(unsigned)((total + 255) / 256), 256, 0, stream>>>(H, ldh, src, dst, M, ldm, E);
    } else if (nfeat == 32) {
        const long total = (long)E * 8;
        gcn_aggregate<8><<<(unsigned)((total + 255) / 256), 256, 0, stream>>>(H, ldh, src, dst, M, ldm, E);
    } else {
        const long total = (long)E * nfeat;
        gcn_aggregate_any<<<(unsigned)((total + 255) / 256), 256, 0, stream>>>(H, ldh, src, dst, M, ldm, nfeat, E);
    }
}

extern "C" void kernel_launch(void* const* d_in, const int* in_sizes, int n_in,
                              void* d_out, int out_size, void* d_ws, size_t ws_size,
                              hipStream_t stream) {
    const float* feats = (const float*)d_in[0];
    const int*   src   = (const int*)d_in[1];
    const int*   dst   = (const int*)d_in[2];
    const float* W0    = (const float*)d_in[3];
    const float* b0    = (const float*)d_in[4];
    const float* W1    = (const float*)d_in[5];
    const float* b1    = (const float*)d_in[6];
    const float* W2    = (const float*)d_in[7];
    const float* b2    = (const float*)d_in[8];
    float* out = (float*)d_out;

    const int h = in_sizes[4];               // 64
    const int d = in_sizes[3] / h;           // 64
    const int c = in_sizes[8];               // 40
    const int N = in_sizes[0] / d;           // 100000
    const int E = in_sizes[1];               // 1,600,000

    const int nctH = (h + 15) / 16;          // 4
    const int nctC = (c + 15) / 16;          // 3
    int ldbuf = h;
    if (nctC * 16 > ldbuf) ldbuf = nctC * 16;

    // Workspace layout (floats): normO[N] | normI[N] | buf0[N*ldbuf] | buf1[N*ldbuf]
    float* normO = (float*)d_ws;
    float* normI = normO + N;
    float* buf0  = normI + N;
    float* buf1  = buf0 + (long)N * ldbuf;
    (void)ws_size; (void)n_in; (void)out_size;

    // ---- degrees & normalization factors (shared by all layers) ----
    gcn_zero<<<2048, 256, 0, stream>>>(normO, (long)2 * N);
    gcn_degrees<<<(unsigned)((E + 255) / 256), 256, 0, stream>>>(src, dst, normO, normI, E);
    gcn_invsqrt<<<(unsigned)(((long)2 * N + 255) / 256), 256, 0, stream>>>(normO, (long)2 * N);

    // ---- layer 0: feats -> buf1 ----
    launch_gemm(nctH, d, feats, d, W0, h, normO, buf0, ldbuf, N, stream);
    gcn_zero<<<4096, 256, 0, stream>>>(buf1, (long)N * h);
    launch_aggregate(buf0, ldbuf, src, dst, buf1, h, h, E, stream);
    gcn_finalize<<<(unsigned)(((long)N * h + 255) / 256), 256, 0, stream>>>(buf1, h, normI, b0, N, 1);

    // ---- layer 1: buf1 -> buf1 ----
    launch_gemm(nctH, h, buf1, h, W1, h, normO, buf0, ldbuf, N, stream);
    gcn_zero<<<4096, 256, 0, stream>>>(buf1, (long)N * h);
    launch_aggregate(buf0, ldbuf, src, dst, buf1, h, h, E, stream);
    gcn_finalize<<<(unsigned)(((long)N * h + 255) / 256), 256, 0, stream>>>(buf1, h, normI, b1, N, 1);

    // ---- layer 2: buf1 -> d_out (no ReLU) ----
    launch_gemm(nctC, h, buf1, h, W2, c, normO, buf0, ldbuf, N, stream);
    gcn_zero<<<4096, 256, 0, stream>>>(out, (long)N * c);
    launch_aggregate(buf0, ldbuf, src, dst, out, c, c, E, stream);
    gcn_finalize<<<(unsigned)(((long)N * c + 255) / 256), 256, 0, stream>>>(out, c, normI, b2, N, 0);
}